// OrdinalLoss_9861244911666
// MI455X (gfx1250) — compile-verified
//
#include <hip/hip_runtime.h>
#include <cstdint>
#include <cstddef>

#define C_DIM 256
#define WAVES_PER_BLOCK 8
#define ROWS_PER_WAVE 4
#define ROWS_PER_BLOCK (WAVES_PER_BLOCK * ROWS_PER_WAVE)

// Issue an async global->LDS copy of one 1KB row (C_DIM floats).
// Each of the 32 lanes copies 16B at lane*16 and 512+lane*16.
__device__ __forceinline__ void async_copy_row(const float* gsrc, unsigned lds_base, int lane) {
  unsigned long long g0 = (unsigned long long)(const void*)gsrc + (unsigned)(lane * 16);
  unsigned l0 = lds_base + (unsigned)(lane * 16);
  asm volatile("global_load_async_to_lds_b128 %0, %1, off"
               :: "v"(l0), "v"(g0) : "memory");
  asm volatile("global_load_async_to_lds_b128 %0, %1, off"
               :: "v"(l0 + 512u), "v"(g0 + 512ull) : "memory");
}

__global__ void __launch_bounds__(256)
ordinal_loss_partial(const float* __restrict__ pred,
                     const int* __restrict__ tgt,
                     float* __restrict__ partials,
                     int nrows) {
  __shared__ __align__(16) float smem[WAVES_PER_BLOCK * 2 * C_DIM]; // 16 KB double-buffer
  __shared__ float wacc[WAVES_PER_BLOCK];

  const int lane = threadIdx.x & 31;
  const int wv   = threadIdx.x >> 5;
  const int gw   = blockIdx.x * WAVES_PER_BLOCK + wv;
  const int row0 = gw * ROWS_PER_WAVE;

  // Raw 32-bit LDS byte offsets (flat addr low 32 bits == LDS offset, ISA 10.2)
  const unsigned lbase0 =
      (unsigned)(unsigned long long)(const void*)&smem[(wv * 2 + 0) * C_DIM];
  const unsigned lbase1 =
      (unsigned)(unsigned long long)(const void*)&smem[(wv * 2 + 1) * C_DIM];

  float acc = 0.0f;

  // Prefetch row 0 into buffer 0 (clamp so loads stay in-bounds; result masked later)
  {
    int r = row0; if (r >= nrows) r = nrows - 1; if (r < 0) r = 0;
    async_copy_row(pred + (size_t)r * C_DIM, lbase0, lane);
  }

#pragma unroll
  for (int k = 0; k < ROWS_PER_WAVE; ++k) {
    if (k + 1 < ROWS_PER_WAVE) {
      int rn = row0 + k + 1; if (rn >= nrows) rn = nrows - 1;
      async_copy_row(pred + (size_t)rn * C_DIM, ((k + 1) & 1) ? lbase1 : lbase0, lane);
      // rows complete in order: <=2 outstanding means row k's 1KB is in LDS
      asm volatile("s_wait_asynccnt 0x2" ::: "memory");
    } else {
      asm volatile("s_wait_asynccnt 0x0" ::: "memory");
    }

    const int r = row0 + k;
    const float* buf = &smem[(wv * 2 + (k & 1)) * C_DIM];

    // lane holds elements j in {4l..4l+3} and {128+4l..128+4l+3}
    float4 x0 = *(const float4*)&buf[lane * 4];
    float4 x1 = *(const float4*)&buf[C_DIM / 2 + lane * 4];
    float v[8] = {x0.x, x0.y, x0.z, x0.w, x1.x, x1.y, x1.z, x1.w};

    // row max (lane-local then wave32 xor-tree)
    float m = v[0];
#pragma unroll
    for (int i = 1; i < 8; ++i) m = fmaxf(m, v[i]);
#pragma unroll
    for (int s = 16; s > 0; s >>= 1) m = fmaxf(m, __shfl_xor(m, s, 32));

    int t = (r < nrows) ? tgt[r] : 0;

    // weighted sums
    float s0 = 0.0f, s1 = 0.0f, s2 = 0.0f, pt = 0.0f;
    const float j0 = (float)(lane * 4);
#pragma unroll
    for (int i = 0; i < 8; ++i) {
      float j = j0 + (float)(i & 3) + ((i >= 4) ? 128.0f : 0.0f);
      float e = __expf(v[i] - m);
      s0 += e;
      s1 = fmaf(e, j, s1);
      s2 = fmaf(e * j, j, s2);
    }
    // target logit: exactly one lane contributes, fold into the sum tree
    {
      int tl = (t & 127) >> 2;
      int ts = (t & 3) + ((t >= 128) ? 4 : 0);
      if (lane == tl) pt = v[ts];
    }
#pragma unroll
    for (int s = 16; s > 0; s >>= 1) {
      s0 += __shfl_xor(s0, s, 32);
      s1 += __shfl_xor(s1, s, 32);
      s2 += __shfl_xor(s2, s, 32);
      pt += __shfl_xor(pt, s, 32);
    }

    if (r < nrows) {
      float tf  = (float)t;
      float ce  = m - pt + __logf(s0);
      float pen = fmaf(tf, fmaf(tf, s0, -2.0f * s1), s2) / s0; // S2 - 2tS1 + t^2 S0 over S0
      acc += ce + pen; // LAMBDA_ORD = 1
    }
  }

  // acc is wave-uniform (built from reduced values)
  if (lane == 0) wacc[wv] = acc;
  __syncthreads();
  if (threadIdx.x == 0) {
    float b = 0.0f;
#pragma unroll
    for (int i = 0; i < WAVES_PER_BLOCK; ++i) b += wacc[i];
    partials[blockIdx.x] = b;
  }
}

__global__ void __launch_bounds__(256)
ordinal_loss_finalize(const float* __restrict__ partials, int n,
                      float* __restrict__ out, float scale) {
  __shared__ float wsum[8];
  float a = 0.0f;
  for (int i = threadIdx.x; i < n; i += blockDim.x) a += partials[i];
#pragma unroll
  for (int s = 16; s > 0; s >>= 1) a += __shfl_xor(a, s, 32);
  const int lane = threadIdx.x & 31;
  const int wv   = threadIdx.x >> 5;
  if (lane == 0) wsum[wv] = a;
  __syncthreads();
  if (threadIdx.x == 0) {
    float t = 0.0f;
#pragma unroll
    for (int i = 0; i < 8; ++i) t += wsum[i];
    out[0] = t * scale;
  }
}

extern "C" void kernel_launch(void* const* d_in, const int* in_sizes, int n_in,
                              void* d_out, int out_size, void* d_ws, size_t ws_size,
                              hipStream_t stream) {
  const float* pred = (const float*)d_in[0];
  const int*   tgt  = (const int*)d_in[1];
  const int    Bn   = in_sizes[1];

  const int nblocks = (Bn + ROWS_PER_BLOCK - 1) / ROWS_PER_BLOCK;
  float* partials = (float*)d_ws;

  ordinal_loss_partial<<<nblocks, 256, 0, stream>>>(pred, tgt, partials, Bn);
  ordinal_loss_finalize<<<1, 256, 0, stream>>>(partials, nblocks, (float*)d_out,
                                               1.0f / (float)Bn);
}